// Flow_47296179864280
// MI455X (gfx1250) — compile-verified
//
#include <hip/hip_runtime.h>
#include <math.h>

// MI455X / gfx1250: wave32, WMMA (not MFMA).
// f32 WMMA 16x16x4 keeps full reference precision for the sequential scan.

typedef __attribute__((ext_vector_type(2))) float v2f;
typedef __attribute__((ext_vector_type(8))) float v8f;

#define N_STATES 512
#define N_IN     32
#define N_OUT    16
#define T_KNOTS  4096
#define N_STEPS  (T_KNOTS - 1)

__device__ __forceinline__ v8f wmma_f32x4(v2f a, v2f b, v8f c) {
  // 8 args: (neg_a, A, neg_b, B, c_mod, C, reuse_a, reuse_b)
  return __builtin_amdgcn_wmma_f32_16x16x4_f32(false, a, false, b, (short)0, c,
                                               false, false);
}

// AxLDS = A * xv, A is 512x512 row-major f32 (L2-resident), xv in LDS.
// Executed cooperatively by all 32 waves; wave w owns output rows [16w,16w+16).
// A-operand (16x4 f32): lane L holds row (L&15), K = kbase + (L>>4)*2 + {0,1}.
// B-operand (4x16 f32): x broadcast to all 16 columns -> lane L holds
// x[kbase + (L>>4)*2 + {0,1}], so every column of D equals A@x.
__device__ __forceinline__ void matvec512(const float* __restrict__ A,
                                          const float* xv, float* Ax,
                                          int lane, int w) {
  const int hi = lane >> 4;
  const float* Arow = A + ((w << 4) + (lane & 15)) * N_STATES + (hi << 1);
  const float* xp = xv + (hi << 1);
  v8f a0 = {}, a1 = {}, a2 = {}, a3 = {};  // 4 independent accumulator chains
#pragma unroll 4
  for (int kc = 0; kc < 128; kc += 4) {
    v2f af, bf;
    af = *(const v2f*)(Arow + ((kc + 0) << 2));
    bf = *(const v2f*)(xp + ((kc + 0) << 2));
    a0 = wmma_f32x4(af, bf, a0);
    af = *(const v2f*)(Arow + ((kc + 1) << 2));
    bf = *(const v2f*)(xp + ((kc + 1) << 2));
    a1 = wmma_f32x4(af, bf, a1);
    af = *(const v2f*)(Arow + ((kc + 2) << 2));
    bf = *(const v2f*)(xp + ((kc + 2) << 2));
    a2 = wmma_f32x4(af, bf, a2);
    af = *(const v2f*)(Arow + ((kc + 3) << 2));
    bf = *(const v2f*)(xp + ((kc + 3) << 2));
    a3 = wmma_f32x4(af, bf, a3);
  }
  v8f acc = (a0 + a1) + (a2 + a3);
  // D layout: VGPR r = row (L>>4)*8 + r; all columns identical -> one lane
  // per half-wave writes its 8 rows to LDS.
  if ((lane & 15) == 0) {
    float* dst = Ax + (w << 4) + (hi << 3);
#pragma unroll
    for (int r = 0; r < 8; ++r) dst[r] = acc[r];
  }
}

__global__ __launch_bounds__(1024) void flow_scan_kernel(
    const float* __restrict__ x0, const float* __restrict__ t,
    const float* __restrict__ uc, const float* __restrict__ A,
    const float* __restrict__ B, float* __restrict__ xs) {
  __shared__ float xLDS[N_STATES];    // current stage input vector
  __shared__ float AxLDS[N_STATES];   // A @ x_stage
  __shared__ float ucLDS[4 * N_IN];   // this step's (d,c,b,a) spline coeffs

  const int tid = threadIdx.x;
  const int lane = tid & 31;
  const int w = tid >> 5;
  const float dt = t[1] - t[0];

  float x = 0.f;
  if (tid < N_STATES) {
    x = x0[tid];
    xLDS[tid] = x;
    xs[tid] = x;  // xs row 0 = x0
  }
  __syncthreads();

  const float ctab[6] = {0.f, 0.2f, 0.3f, 0.8f, 8.f / 9.f, 1.f};
  float kk0 = 0.f, kk1 = 0.f, kk2 = 0.f, kk3 = 0.f, kk4 = 0.f, kk5 = 0.f;

  for (int step = 0; step < N_STEPS; ++step) {
    // Stage the 4x32 spline coefficients for this interval into LDS.
    if (tid < 4 * N_IN)
      ucLDS[tid] = uc[(tid >> 5) * (N_STEPS * N_IN) + step * N_IN + lane];
    __syncthreads();

    // Bq = B @ {d,c,b,a} per-element: 128 FMAs off the matrix pipe.
    float Bq0 = 0.f, Bq1 = 0.f, Bq2 = 0.f, Bq3 = 0.f;
    if (tid < N_STATES) {
      const float* Brow = B + tid * N_IN;
#pragma unroll
      for (int j = 0; j < N_IN; ++j) {
        float bj = Brow[j];
        Bq0 = fmaf(bj, ucLDS[j], Bq0);            // B@d
        Bq1 = fmaf(bj, ucLDS[N_IN + j], Bq1);     // B@c
        Bq2 = fmaf(bj, ucLDS[2 * N_IN + j], Bq2); // B@b
        Bq3 = fmaf(bj, ucLDS[3 * N_IN + j], Bq3); // B@a
      }
    }

#pragma unroll
    for (int s = 0; s < 6; ++s) {
      matvec512(A, xLDS, AxLDS, lane, w);  // all waves, full EXEC
      __syncthreads();
      if (tid < N_STATES) {
        const float sv = ctab[s] * dt;
        const float bu = fmaf(sv, fmaf(sv, fmaf(sv, Bq0, Bq1), Bq2), Bq3);
        const float ks = tanhf(AxLDS[tid]) + bu;  // k_{s+1}
        if (s == 0) kk0 = ks;
        else if (s == 1) kk1 = ks;
        else if (s == 2) kk2 = ks;
        else if (s == 3) kk3 = ks;
        else if (s == 4) kk4 = ks;
        else kk5 = ks;
        float xn;
        if (s == 0)
          xn = x + dt * (0.2f * kk0);
        else if (s == 1)
          xn = x + dt * (3.f / 40 * kk0 + 9.f / 40 * kk1);
        else if (s == 2)
          xn = x + dt * (44.f / 45 * kk0 - 56.f / 15 * kk1 + 32.f / 9 * kk2);
        else if (s == 3)
          xn = x + dt * (19372.f / 6561 * kk0 - 25360.f / 2187 * kk1 +
                         64448.f / 6561 * kk2 - 212.f / 729 * kk3);
        else if (s == 4)
          xn = x + dt * (9017.f / 3168 * kk0 - 355.f / 33 * kk1 +
                         46732.f / 5247 * kk2 + 49.f / 176 * kk3 -
                         5103.f / 18656 * kk4);
        else
          xn = x + dt * (35.f / 384 * kk0 + 500.f / 1113 * kk2 +
                         125.f / 192 * kk3 - 2187.f / 6784 * kk4 +
                         11.f / 84 * kk5);
        xLDS[tid] = xn;  // next stage input (or next step's x)
        if (s == 5) {
          x = xn;
          xs[(size_t)(step + 1) * N_STATES + tid] = xn;
        }
      }
      __syncthreads();
    }
  }
}

// ys = xs @ C^T : M=4096 (time), N=16 (outputs), K=512. One 16-row tile per
// wave, f32 WMMA chained over 128 k-chunks with 2 accumulator chains.
__global__ __launch_bounds__(256) void ys_gemm_kernel(
    const float* __restrict__ xs, const float* __restrict__ C,
    float* __restrict__ ys) {
  const int lane = threadIdx.x & 31;
  const int wv = threadIdx.x >> 5;
  const int mt = blockIdx.x * 8 + wv;  // tile of 16 time rows, 0..255
  const int hi = lane >> 4;
  const float* Xrow = xs + (size_t)((mt << 4) + (lane & 15)) * N_STATES + (hi << 1);
  const float* Crow = C + (size_t)(lane & 15) * N_STATES + (hi << 1);  // B op: C^T
  v8f a0 = {}, a1 = {};
#pragma unroll 4
  for (int kc = 0; kc < 128; kc += 2) {
    v2f af, bf;
    af = *(const v2f*)(Xrow + ((kc + 0) << 2));
    bf = *(const v2f*)(Crow + ((kc + 0) << 2));
    a0 = wmma_f32x4(af, bf, a0);
    af = *(const v2f*)(Xrow + ((kc + 1) << 2));
    bf = *(const v2f*)(Crow + ((kc + 1) << 2));
    a1 = wmma_f32x4(af, bf, a1);
  }
  v8f acc = a0 + a1;
  // D: lane col = lane&15, rows = hi*8 + r
  float* out = ys + ((size_t)(mt << 4) + (hi << 3)) * N_OUT + (lane & 15);
#pragma unroll
  for (int r = 0; r < 8; ++r) out[r * N_OUT] = acc[r];
}

extern "C" void kernel_launch(void* const* d_in, const int* in_sizes, int n_in,
                              void* d_out, int out_size, void* d_ws,
                              size_t ws_size, hipStream_t stream) {
  (void)in_sizes; (void)n_in; (void)out_size; (void)d_ws; (void)ws_size;
  const float* x0 = (const float*)d_in[0];
  const float* t  = (const float*)d_in[1];
  const float* uc = (const float*)d_in[2];
  const float* A  = (const float*)d_in[3];
  const float* B  = (const float*)d_in[4];
  const float* C  = (const float*)d_in[5];
  float* xs = (float*)d_out;
  float* ys = xs + (size_t)T_KNOTS * N_STATES;

  flow_scan_kernel<<<dim3(1), dim3(1024), 0, stream>>>(x0, t, uc, A, B, xs);
  ys_gemm_kernel<<<dim3(32), dim3(256), 0, stream>>>(xs, C, ys);
}